// WENONetwork_11407433138414
// MI455X (gfx1250) — compile-verified
//
#include <hip/hip_runtime.h>

// Black-Scholes WENO stepper recast as 26 applications of a banded affine
// operator u' = C u + b(t), executed with V_WMMA_F32_16X16X4_F32 on a single
// wave32.  C (81x81, padded to 96x96) lives in LDS; its 29 banded 16x4 A-tiles
// are hoisted to VGPRs before the time loop, so the 26 dependent steps touch
// only LDS (u vector) and the matrix cores.

typedef __attribute__((ext_vector_type(2))) float v2f;
typedef __attribute__((ext_vector_type(8))) float v8f;

#define MG      80          // M in reference
#define NP      96          // padded matrix dimension (6x16)
#define NSTEPS  26          // ceil(T / (0.4 H^2 / sigma^2)) = ceil(25.6)

__global__ __launch_bounds__(32)
void bs_weno_wmma_kernel(const float* __restrict__ w5,   // omegas5 (80,3)
                         const float* __restrict__ w6,   // omegas6 (79,3)
                         float* __restrict__ out)        // (81,)
{
    __shared__ float Cmat[NP * NP];
    __shared__ float u[NP];
    __shared__ float un[NP];

    const int   lane = threadIdx.x;          // 0..31
    const float Hh   = 0.09375f;             // (1.5 - (-6)) / 80, exact
    const float DT   = 1.0f / 26.0f;
    const float SIG2 = 0.09f;
    const float ADV  = 0.1f - 0.5f * SIG2;   // r - sigma^2/2 = 0.055
    const float RATE = 0.1f;
    const float k1   = DT * ADV / Hh;        // flux-difference scale
    const float kx   = DT * 0.5f * SIG2 / (Hh * Hh);  // diffusion scale

    // ---- zero C, init u0 (payoff) and un --------------------------------
    for (int idx = lane; idx < NP * NP; idx += 32) Cmat[idx] = 0.0f;
    for (int idx = lane; idx < NP; idx += 32) {
        float v = 0.0f;
        if (idx <= MG) {
            float x = -6.0f + (float)idx * Hh;
            v = fmaxf(50.0f * __expf(x) - 50.0f, 0.0f);
        }
        u[idx]  = v;
        un[idx] = 0.0f;
    }
    __syncthreads();

    // ---- build step matrix C: rows 1..79 (rows 0 and 80 stay zero) ------
    for (int i = 1 + lane; i <= MG - 1; i += 32) {
        const int j = i - 1;
        const float inv6 = 1.0f / 6.0f;
        float w50a = w5[3*j+0],     w50b = w5[3*j+1],     w50c = w5[3*j+2];
        float w51a = w5[3*(j+1)+0], w51b = w5[3*(j+1)+1], w51c = w5[3*(j+1)+2];
        float w6a  = w6[3*j+0],     w6b  = w6[3*j+1],     w6c  = w6[3*j+2];

        // uhat_j coefficients over up[j .. j+4]
        float aj[5]  = { 2.0f*w50a*inv6, (-7.0f*w50a - w50b)*inv6,
                         (11.0f*w50a + 5.0f*w50b + 2.0f*w50c)*inv6,
                         (2.0f*w50b + 5.0f*w50c)*inv6, -w50c*inv6 };
        float aj1[5] = { 2.0f*w51a*inv6, (-7.0f*w51a - w51b)*inv6,
                         (11.0f*w51a + 5.0f*w51b + 2.0f*w51c)*inv6,
                         (2.0f*w51b + 5.0f*w51c)*inv6, -w51c*inv6 };
        // uxx coefficients over up[j+1 .. j+5]
        float cx[5]  = { w6a, -2.0f*w6a + w6b, w6a - 2.0f*w6b + w6c,
                         w6b - 2.0f*w6c, w6c };

        float cup[6] = {0.f, 0.f, 0.f, 0.f, 0.f, 0.f};   // columns up[j..j+5]
        #pragma unroll
        for (int q = 0; q < 5; ++q) cup[q]     -= k1 * aj[q];
        #pragma unroll
        for (int q = 0; q < 5; ++q) cup[q + 1] += k1 * aj1[q];
        #pragma unroll
        for (int p = 0; p < 5; ++p) cup[p + 1] += kx * cx[p];

        // map ghost-extended columns up[] -> u[] and scatter into row i
        float* row = &Cmat[i * NP];
        #pragma unroll
        for (int q = 0; q < 6; ++q) {
            int jc = j + q;
            float w = cup[q];
            if (jc <= 1)        row[0]       += w;            // up[0,1] = u[0]
            else if (jc <= 82)  row[jc - 2]  += w;            // up[k] = u[k-2]
            else {              row[79] -= w; row[80] += 2.0f * w; } // up[83]=2u80-u79
        }
        row[i] += 1.0f - DT * RATE;                           // identity + decay
    }
    __syncthreads();

    // ---- hoist the 29 banded A-tiles of C into VGPRs --------------------
    // tile tt (rows 16tt..16tt+15) needs k-chunks [max(0,4tt-1) .. 4tt+4]
    v2f Areg[29];
    {
        int cnt = 0;
        #pragma unroll
        for (int tt = 0; tt < 5; ++tt) {
            const int cmin = (tt == 0) ? 0 : 4 * tt - 1;
            #pragma unroll
            for (int ch = cmin; ch <= 4 * tt + 4; ++ch) {
                int m  = tt * 16 + (lane & 15);
                int kk = ch * 4 + ((lane < 16) ? 0 : 2);
                Areg[cnt++] = *(const v2f*)&Cmat[m * NP + kk];
            }
        }
    }

    // ---- time loop: 26 x (u' = C u;  u'[80] = bc(t)) --------------------
    const float s_right = 50.0f * __expf(1.5f);
    float t = 0.0f;
    for (int s = 0; s < NSTEPS; ++s) {
        int cnt = 0;
        #pragma unroll
        for (int tt = 0; tt < 5; ++tt) {
            v8f acc = {0.f, 0.f, 0.f, 0.f, 0.f, 0.f, 0.f, 0.f};
            const int cmin = (tt == 0) ? 0 : 4 * tt - 1;
            #pragma unroll
            for (int ch = cmin; ch <= 4 * tt + 4; ++ch) {
                int kk = ch * 4 + ((lane < 16) ? 0 : 2);
                v2f b = *(const v2f*)&u[kk];     // u[k] replicated over N
                acc = __builtin_amdgcn_wmma_f32_16x16x4_f32(
                          false, Areg[cnt], false, b,
                          (short)0, acc, false, false);
                ++cnt;
            }
            // column N=0 of D: lane 0 holds rows 0..7, lane 16 rows 8..15
            if (lane == 0) {
                #pragma unroll
                for (int r = 0; r < 8; ++r) un[tt * 16 + r] = acc[r];
            } else if (lane == 16) {
                #pragma unroll
                for (int r = 0; r < 8; ++r) un[tt * 16 + 8 + r] = acc[r];
            }
        }
        t += DT;
        if (lane == 0) {
            un[80] = s_right - 50.0f * __expf(-0.1f * t);  // Dirichlet right BC
            un[0]  = 0.0f;
        }
        __syncthreads();
        #pragma unroll
        for (int idx = lane; idx < NP; idx += 32) u[idx] = un[idx];
        __syncthreads();
    }

    for (int idx = lane; idx <= MG; idx += 32) out[idx] = u[idx];
}

extern "C" void kernel_launch(void* const* d_in, const int* in_sizes, int n_in,
                              void* d_out, int out_size, void* d_ws, size_t ws_size,
                              hipStream_t stream) {
    const float* w5 = (const float*)d_in[0];   // omegas5 (80,3) f32
    const float* w6 = (const float*)d_in[1];   // omegas6 (79,3) f32
    float* out = (float*)d_out;                // 81 f32
    (void)in_sizes; (void)n_in; (void)out_size; (void)d_ws; (void)ws_size;
    bs_weno_wmma_kernel<<<1, 32, 0, stream>>>(w5, w6, out);
}